// TemporalAttnLayer_57818849738849
// MI455X (gfx1250) — compile-verified
//
#include <hip/hip_runtime.h>
#include <math.h>

// Problem dims (fixed by the reference)
#define Bn 64
#define Tn 256
#define Mn 512
#define Hn 256

typedef __attribute__((ext_vector_type(2))) float v2f;
typedef __attribute__((ext_vector_type(8))) float v8f;

// ---------------------------------------------------------------------------
// FP32 WMMA GEMM:  C[16m,16n] tiles of  C = A @ W^T (+bias0 +bias1 + ytil*wih)
//   A: row-major [rows, K]   stride lda
//   W: row-major [N, K]      stride ldw   (so B-matrix[k][n] = W[n][k])
//   C: row-major             stride ldc
// One wave (32 lanes) per 16x16 tile; K consumed 4 at a time via
// V_WMMA_F32_16X16X4_F32.  A-tile (16x4 f32): lanes 0-15 hold K=0,1 (float2),
// lanes 16-31 hold K=2,3.  B-tile (4x16) mirrors the K split with N = lane%16.
// ---------------------------------------------------------------------------
__global__ __launch_bounds__(32) void wmma_f32_gemm(
    const float* __restrict__ A, int lda,
    const float* __restrict__ W, int ldw,
    float* __restrict__ C, int ldc, int K,
    const float* __restrict__ bias0,  // [N] or null
    const float* __restrict__ bias1,  // [N] or null
    const float* __restrict__ wih,    // [N] or null  (adds ytil[row]*wih[col])
    const float* __restrict__ ytil)   // [rows] or null
{
  const int tm   = blockIdx.x;      // tile row (16 rows of C)
  const int tn   = blockIdx.y;      // tile col (16 cols of C)
  const int lane = threadIdx.x;
  const int half = lane >> 4;       // 0: K pair {0,1}, 1: K pair {2,3}
  const int l16  = lane & 15;

  const float* Ap = A + (size_t)(tm * 16 + l16) * lda + half * 2;
  const float* Wp = W + (size_t)(tn * 16 + l16) * ldw + half * 2;

  v8f acc = {};
  for (int k = 0; k < K; k += 4) {
    v2f a = *(const v2f*)(Ap + k);
    v2f b = *(const v2f*)(Wp + k);
    acc = __builtin_amdgcn_wmma_f32_16x16x4_f32(
        /*neg_a=*/false, a, /*neg_b=*/false, b,
        /*c_mod=*/(short)0, acc, /*reuse_a=*/false, /*reuse_b=*/false);
  }

  const int col = tn * 16 + l16;
  float badd = 0.f;
  if (bias0) badd += bias0[col];
  if (bias1) badd += bias1[col];
  const float wv = wih ? wih[col] : 0.f;

#pragma unroll
  for (int r = 0; r < 8; ++r) {
    const int row = tm * 16 + r + half * 8;   // C layout: VGPR r -> M=r / r+8
    float v = acc[r] + badd;
    if (wih) v += ytil[row] * wv;
    C[(size_t)row * ldc + col] = v;
  }
}

// ---------------------------------------------------------------------------
// l_it[b,t] = sum_m tanh(x1[b,m] + y1[b,t,m]) * vd[m]      (one block per bt)
// ---------------------------------------------------------------------------
__global__ __launch_bounds__(256) void attn_logits(
    const float* __restrict__ x1, const float* __restrict__ y1,
    const float* __restrict__ vd, float* __restrict__ lit)
{
  const int bt  = blockIdx.x;
  const int b   = bt >> 8;          // T = 256
  const int tid = threadIdx.x;
  float p = 0.f;
  for (int m = tid; m < Mn; m += 256)
    p += tanhf(x1[b * Mn + m] + y1[(size_t)bt * Mn + m]) * vd[m];
  __shared__ float red[256];
  red[tid] = p; __syncthreads();
  for (int s = 128; s > 0; s >>= 1) {
    if (tid < s) red[tid] += red[tid + s];
    __syncthreads();
  }
  if (tid == 0) lit[bt] = red[0];
}

// ---------------------------------------------------------------------------
// Per-batch: softmax over T, context c_t, accumulate output, y_tilda scalar.
// One block (256 threads) per b.
// ---------------------------------------------------------------------------
__global__ __launch_bounds__(256) void softmax_ctx(
    const float* __restrict__ lit, const float* __restrict__ enc,
    const float* __restrict__ y, const float* __restrict__ wt_w,
    const float* __restrict__ wt_b, float* __restrict__ cbuf,
    float* __restrict__ out, float* __restrict__ ytil, int step)
{
  const int b = blockIdx.x, tid = threadIdx.x;
  __shared__ float sh[256];
  __shared__ float beta[Tn];

  const float lv = lit[b * Tn + tid];
  sh[tid] = lv; __syncthreads();
  for (int s = 128; s > 0; s >>= 1) {
    if (tid < s) sh[tid] = fmaxf(sh[tid], sh[tid + s]);
    __syncthreads();
  }
  const float mx = sh[0]; __syncthreads();
  const float e = __expf(lv - mx);
  sh[tid] = e; __syncthreads();
  for (int s = 128; s > 0; s >>= 1) {
    if (tid < s) sh[tid] += sh[tid + s];
    __syncthreads();
  }
  beta[tid] = e / sh[0];
  __syncthreads();

  // c_t[m] = sum_t beta[t]*enc[b,t,m] ; also accumulate into out
  for (int m = tid; m < Mn; m += 256) {
    const float* ep = enc + (size_t)b * Tn * Mn + m;
    float acc = 0.f;
#pragma unroll 4
    for (int t2 = 0; t2 < Tn; ++t2) acc += beta[t2] * ep[(size_t)t2 * Mn];
    cbuf[b * Mn + m] = acc;
    out[b * Mn + m] += acc;
  }
  __syncthreads();

  // y_tilda[b] = c_t . wt_w[0:M] + y[b,step]*wt_w[M] + wt_b
  float p = 0.f;
  for (int m = tid; m < Mn; m += 256) p += cbuf[b * Mn + m] * wt_w[m];
  sh[tid] = p; __syncthreads();
  for (int s = 128; s > 0; s >>= 1) {
    if (tid < s) sh[tid] += sh[tid + s];
    __syncthreads();
  }
  if (tid == 0)
    ytil[b] = sh[0] + y[b * Tn + step] * wt_w[Mn] + wt_b[0];
}

// ---------------------------------------------------------------------------
// LSTM pointwise update of ds = [d | s]  (gate order i,f,g,o)
// ---------------------------------------------------------------------------
__device__ __forceinline__ float sigm(float x) { return 1.f / (1.f + __expf(-x)); }

__global__ __launch_bounds__(256) void lstm_update(
    const float* __restrict__ gates, float* __restrict__ ds)
{
  const int b = blockIdx.x, h = threadIdx.x;
  const float gi = gates[b * 4 * Hn + 0 * Hn + h];
  const float gf = gates[b * 4 * Hn + 1 * Hn + h];
  const float gg = gates[b * 4 * Hn + 2 * Hn + h];
  const float go = gates[b * 4 * Hn + 3 * Hn + h];
  const float s_old = ds[b * 2 * Hn + Hn + h];
  const float s_new = sigm(gf) * s_old + sigm(gi) * tanhf(gg);
  ds[b * 2 * Hn + h]      = sigm(go) * tanhf(s_new);
  ds[b * 2 * Hn + Hn + h] = s_new;
}

__global__ void zero_init(float* __restrict__ out, float* __restrict__ ds) {
  const int i = blockIdx.x * blockDim.x + threadIdx.x;
  if (i < Bn * Mn) out[i] = 0.f;          // B*M == B*2H == 32768
  if (i < Bn * 2 * Hn) ds[i] = 0.f;
}

extern "C" void kernel_launch(void* const* d_in, const int* in_sizes, int n_in,
                              void* d_out, int out_size, void* d_ws, size_t ws_size,
                              hipStream_t stream) {
  const float* enc  = (const float*)d_in[0];   // [B,T,M]
  const float* y    = (const float*)d_in[1];   // [B,T,1]
  const float* Wd_w = (const float*)d_in[2];   // [M,2H]
  const float* Wd_b = (const float*)d_in[3];   // [M]
  const float* Ud_w = (const float*)d_in[4];   // [M,M]
  const float* vd_w = (const float*)d_in[5];   // [1,M]
  const float* wt_w = (const float*)d_in[6];   // [1,M+1]
  const float* wt_b = (const float*)d_in[7];   // [1]
  const float* Wih  = (const float*)d_in[8];   // [4H,1]
  const float* Whh  = (const float*)d_in[9];   // [4H,H]
  const float* bih  = (const float*)d_in[10];  // [4H]
  const float* bhh  = (const float*)d_in[11];  // [4H]
  float* out = (float*)d_out;                  // [B,M]

  float* ws    = (float*)d_ws;
  float* y1    = ws;                               // B*T*M  (33.5 MB)
  float* ds    = y1 + (size_t)Bn * Tn * Mn;        // B*2H
  float* x1    = ds + Bn * 2 * Hn;                 // B*M
  float* lit   = x1 + Bn * Mn;                     // B*T
  float* cbuf  = lit + Bn * Tn;                    // B*M
  float* ytil  = cbuf + Bn * Mn;                   // B
  float* gates = ytil + Bn;                        // B*4H

  zero_init<<<128, 256, 0, stream>>>(out, ds);

  // Hoisted: y1 = enc @ Ud^T   ([B*T, M] x [M, M])
  wmma_f32_gemm<<<dim3(Bn * Tn / 16, Mn / 16), 32, 0, stream>>>(
      enc, Mn, Ud_w, Mn, y1, Mn, Mn, nullptr, nullptr, nullptr, nullptr);

  for (int t = 0; t < Tn; ++t) {
    // x1 = [d,s] @ Wd_w^T + Wd_b   ([B, 2H] x [2H, M])
    wmma_f32_gemm<<<dim3(Bn / 16, Mn / 16), 32, 0, stream>>>(
        ds, 2 * Hn, Wd_w, 2 * Hn, x1, Mn, 2 * Hn, Wd_b, nullptr, nullptr, nullptr);

    attn_logits<<<Bn * Tn, 256, 0, stream>>>(x1, y1, vd_w, lit);

    softmax_ctx<<<Bn, 256, 0, stream>>>(lit, enc, y, wt_w, wt_b,
                                        cbuf, out, ytil, t);

    // gates = d @ Whh^T + ytil*Wih + bih + bhh   ([B, H] x [H, 4H])
    wmma_f32_gemm<<<dim3(Bn / 16, 4 * Hn / 16), 32, 0, stream>>>(
        ds, 2 * Hn, Whh, Hn, gates, 4 * Hn, Hn, bih, bhh, Wih, ytil);

    lstm_update<<<Bn, 256, 0, stream>>>(gates, ds);
  }
}